// VoxelsToImage_41704132444292
// MI455X (gfx1250) — compile-verified
//
#include <hip/hip_runtime.h>
#include <hip/hip_bf16.h>
#include <stdint.h>
#include <math.h>

// Geometry fixed by the reference setup.
#define W_DIM 128
#define L_DIM 128
#define H_DIM 64
#define N_VOX (W_DIM * L_DIM * H_DIM)   // 1 << 20
#define C_DIM 32
#define BLK   256
#define BLOCKS_PER_BATCH (N_VOX / BLK)  // 4096 = 1 << 12

#define VOXEL_SIZE 0.05f
#define ORG_X (-3.2f)
#define ORG_Y (-3.2f)
#define ORG_Z (0.0f)

__global__ __launch_bounds__(BLK) void vox2img_zero(float4* __restrict__ out4, int n4) {
    int i = blockIdx.x * BLK + threadIdx.x;
    if (i < n4) out4[i] = make_float4(0.f, 0.f, 0.f, 0.f);
}

__global__ __launch_bounds__(BLK) void vox2img_scatter(
    const float* __restrict__ voxel,   // (b, 32, N)
    const int*   __restrict__ occ,     // (b, 1, N)
    const float* __restrict__ extr,    // (b, 4, 4) row-major
    const int* __restrict__ ih_p,
    const int* __restrict__ iw_p,
    const int* __restrict__ hfov_p,
    unsigned int* __restrict__ out)    // (b, 32, ih*iw) viewed as u32 bits
{
    const int b = blockIdx.x >> 12;                              // block / BLOCKS_PER_BATCH
    const int n = ((blockIdx.x & (BLOCKS_PER_BATCH - 1)) << 8) | threadIdx.x;

    // ---- gfx1250 async prefetch: occupancy -> LDS, overlapped with projection math.
    // Per-lane global address -> per-lane LDS slot; each lane later reads only its
    // own slot, so s_wait_asynccnt 0 is sufficient (no cross-wave sharing).
    __shared__ int s_occ[BLK];
    {
        unsigned lds_off = (unsigned)(uintptr_t)(&s_occ[threadIdx.x]);
        uint64_t gaddr   = (uint64_t)(uintptr_t)(occ + ((size_t)b << 20) + n);
        asm volatile("global_load_async_to_lds_b32 %0, %1, off"
                     :: "v"(lds_off), "v"(gaddr)
                     : "memory");
    }

    // ---- voxel centroid
    const int x = n >> 13;          // / (L*H)
    const int y = (n >> 6) & 127;   // / H % L
    const int z = n & 63;           // % H
    const float px = ORG_X + ((float)x + 0.5f) * VOXEL_SIZE;
    const float py = ORG_Y + ((float)y + 0.5f) * VOXEL_SIZE;
    const float pz = ORG_Z + ((float)z + 0.5f) * VOXEL_SIZE;

    // ---- camera transform (extrinsics uniform per block -> scalar loads)
    const float* E = extr + b * 16;
    const float cx = E[0] * px + E[1] * py + E[2]  * pz + E[3];
    const float cy = E[4] * px + E[5] * py + E[6]  * pz + E[7];
    const float cz = E[8] * px + E[9] * py + E[10] * pz + E[11];

    const int   ih   = ih_p[0];
    const int   iw   = iw_p[0];
    const float hfov = (float)hfov_p[0];
    // Match reference order exactly: rad = deg2rad(hfov); focal = iw/2 / tan(rad/2).
    // Precise tanf (OCML, ~1 ulp) -- a fast-approx tan here shifts focal enough to
    // flip truncated pixel indices for boundary voxels.
    const float rad   = hfov * 0.017453292519943295f;   // pi/180
    const float focal = (0.5f * (float)iw) / tanf(rad * 0.5f);

    const float rowf = focal * cy / cz + 0.5f * (float)ih;
    const float colf = focal * cx / cz + 0.5f * (float)iw;

    const bool in_bounds = (cz > 0.0f) &
                           (rowf > 0.0f) & (rowf < (float)ih) &
                           (colf > 0.0f) & (colf < (float)iw);

    // ---- consume prefetched occupancy
    asm volatile("s_wait_asynccnt 0" ::: "memory");
    const int occ_v = s_occ[threadIdx.x];

    if (in_bounds & (occ_v > 0)) {
        const int row_i = (int)rowf;
        const int col_i = (int)colf;
        const int flat  = col_i * iw + row_i;
        const size_t img = (size_t)ih * (size_t)iw;

        const float*  vptr = voxel + (size_t)b * C_DIM * (size_t)N_VOX + (size_t)n;
        unsigned int* optr = out   + (size_t)b * C_DIM * img + (size_t)flat;

        // 32 coalesced streaming loads (non-temporal: single-use data, keep the
        // 64MB output resident in L2 for the atomic RMWs), then fire-and-forget
        // u32 atomic max (exact for clamped-nonnegative floats vs 0-init).
        #pragma unroll
        for (int c = 0; c < C_DIM; ++c) {
            float v = __builtin_nontemporal_load(vptr + (size_t)c * N_VOX);
            v = fmaxf(v, 0.0f);
            if (v > 0.0f) {
                atomicMax(optr + (size_t)c * img, __float_as_uint(v));
            }
        }
    }
}

extern "C" void kernel_launch(void* const* d_in, const int* in_sizes, int n_in,
                              void* d_out, int out_size, void* d_ws, size_t ws_size,
                              hipStream_t stream) {
    const float* voxel  = (const float*)d_in[0];
    const int*   occ    = (const int*)  d_in[1];
    const float* extr   = (const float*)d_in[2];
    const int*   ih_p   = (const int*)  d_in[3];
    const int*   iw_p   = (const int*)  d_in[4];
    const int*   hfov_p = (const int*)  d_in[5];

    const int b = in_sizes[1] / N_VOX;   // occupancy = (b, 1, N)

    // zero-init output (atomic-max identity); out_size divisible by 4
    const int n4 = out_size / 4;
    vox2img_zero<<<(n4 + BLK - 1) / BLK, BLK, 0, stream>>>((float4*)d_out, n4);

    vox2img_scatter<<<b * BLOCKS_PER_BATCH, BLK, 0, stream>>>(
        voxel, occ, extr, ih_p, iw_p, hfov_p, (unsigned int*)d_out);
}